// ImageParser_68745246540061
// MI455X (gfx1250) — compile-verified
//
#include <hip/hip_runtime.h>

// CDNA5 (gfx1250) — wave32, WMMA f32 16x16x4.
typedef float v2f __attribute__((ext_vector_type(2)));
typedef float v8f __attribute__((ext_vector_type(8)));

#define NB   256          // batch
#define DD   128          // proj dim (M of GEMM)
#define NP   1296         // pixels  (K of GEMM)
#define NC   64           // classes (N of GEMM)
#define EPSF 1e-4f

__global__ __launch_bounds__(256)
void ImageParser_68745246540061_kernel(const float* __restrict__ x,
                                       const int*   __restrict__ cm,
                                       float*       __restrict__ out) {
    __shared__ int   s_cm[NP];          // per-image class map
    __shared__ float s_cnt[NC];         // per-class pixel counts
    __shared__ float s_mean[DD * NC];   // sums -> means (in place)

    const int tid  = threadIdx.x;
    const int b    = blockIdx.x;
    const int lane = tid & 31;
    const int wv   = tid >> 5;          // 8 waves, one M-tile each

    // ---- phase 0: stage class map, compute counts (ds_add_f32 atomics) ----
    const int* cmb = cm + b * NP;
    for (int i = tid; i < NP; i += 256) s_cm[i] = cmb[i];
    if (tid < NC) s_cnt[tid] = 0.0f;
    __syncthreads();
    for (int i = tid; i < NP; i += 256) atomicAdd(&s_cnt[s_cm[i]], 1.0f);
    __syncthreads();

    // ---- phase 1: sums[128x64] = X[128x1296] * onehot[1296x64] via WMMA ----
    // A frag (16x4 f32): lane L -> row M = wv*16 + (L&15); K pair kb = (L<16?0:2)
    const int  mrow = wv * 16 + (lane & 15);
    const int  kb   = (lane >> 4) << 1;            // 0 or 2
    const float* xrow = x + ((size_t)b * DD + mrow) * NP;
    const int  myc  = lane & 15;                   // B/D column within tile

    v8f acc[4] = {};                               // 4 N-tiles of 16 classes

    for (int n0 = 0; n0 < NP; n0 += 4) {
        // 8-byte aligned vector load: K=kb, kb+1 for this lane's row
        v2f a = *(const v2f*)(xrow + n0 + kb);
        // class ids for the two K rows this lane supplies to B
        int c0 = s_cm[n0 + kb];
        int c1 = s_cm[n0 + kb + 1];
#pragma unroll
        for (int t = 0; t < 4; ++t) {
            int cc = t * 16 + myc;
            v2f bf;
            bf.x = (c0 == cc) ? 1.0f : 0.0f;
            bf.y = (c1 == cc) ? 1.0f : 0.0f;
            // 8 args: (neg_a, A, neg_b, B, c_mod, C, reuse_a, reuse_b)
            acc[t] = __builtin_amdgcn_wmma_f32_16x16x4_f32(
                false, a, false, bf, (short)0, acc[t], false, false);
        }
    }

    // C/D layout: VGPR j, lanes 0-15 -> M=j, lanes 16-31 -> M=j+8
    const int rbase = wv * 16 + ((lane >> 4) << 3);
#pragma unroll
    for (int t = 0; t < 4; ++t) {
        int col = t * 16 + myc;
#pragma unroll
        for (int j = 0; j < 8; ++j)
            s_mean[(rbase + j) * NC + col] = acc[t][j];
    }
    __syncthreads();

    // ---- phase 1.5: means = sums / (counts + eps), in place ----
    for (int i = tid; i < DD * NC; i += 256)
        s_mean[i] = s_mean[i] / (s_cnt[i & (NC - 1)] + EPSF);
    __syncthreads();

    // ---- phase 2: per-pixel mean squared distance to class mean ----
    // For fixed d, lanes read consecutive n -> fully coalesced global loads.
    const float* xb = x + (size_t)b * DD * NP;
    for (int n = tid; n < NP; n += 256) {
        int   c    = s_cm[n];
        float accd = 0.0f;
#pragma unroll 4
        for (int d = 0; d < DD; ++d) {
            float v    = xb[(size_t)d * NP + n];
            float m    = s_mean[d * NC + c];
            float diff = v - m;
            accd = fmaf(diff, diff, accd);
        }
        out[(size_t)b * NP + n] = accd * (1.0f / (float)DD);
    }
}

extern "C" void kernel_launch(void* const* d_in, const int* in_sizes, int n_in,
                              void* d_out, int out_size, void* d_ws, size_t ws_size,
                              hipStream_t stream) {
    (void)in_sizes; (void)n_in; (void)d_ws; (void)ws_size; (void)out_size;
    const float* x  = (const float*)d_in[0];   // [B, D, HW, HW] f32
    const int*   cm = (const int*)d_in[1];     // [B, HW, HW]   i32
    float*       o  = (float*)d_out;           // [B, N, 1]     f32
    ImageParser_68745246540061_kernel<<<NB, 256, 0, stream>>>(x, cm, o);
}